// ParallelPhysicalRecurrentLayer_81741817577758
// MI455X (gfx1250) — compile-verified
//
#include <hip/hip_runtime.h>
#include <stdint.h>

// Problem constants (from reference): B,L,R,C,H,Wf = 2,24,8,32,48,49
#define B_   2
#define L_   24
#define R_   8
#define C_   32
#define H_   48
#define W_   49
#define RC_  (R_ * C_)            // 256
#define HW_  (H_ * W_)            // 2352
#define LW_  (L_ * W_)            // 1176 coefficients per (b,rc) tile
#define NA_  (B_ * L_ * RC_ * W_) // 602112   A-coefficient count
#define NTOT_ ((size_t)B_ * L_ * RC_ * HW_)   // 28901376 elements per plane
#define HW_CHUNK_ 784             // 2352 / 3 exactly
#define LSTRIDE_ ((size_t)RC_ * HW_)          // element stride between l-steps

typedef uint32_t u32x4 __attribute__((ext_vector_type(4)));
typedef int32_t  i32x8 __attribute__((ext_vector_type(8)));
typedef int32_t  i32x4 __attribute__((ext_vector_type(4)));

#if defined(__has_builtin)
#if __has_builtin(__builtin_amdgcn_tensor_load_to_lds) && __has_builtin(__builtin_amdgcn_s_wait_tensorcnt)
#define HAVE_TDM 1
#endif
#endif
#ifndef HAVE_TDM
#define HAVE_TDM 0
#endif

// ---------------------------------------------------------------------------
// Phase 1: A[b][l][rc][w] = exp(-nu*dt) * (cos(th*dt), sin(th*dt))
// Planar layout in workspace: Ar plane [NA_] followed by Ai plane [NA_].
// Removes the 48x (H) redundancy of exp/sincos from the hot kernel.
// ---------------------------------------------------------------------------
__global__ __launch_bounds__(256) void
compute_A_kernel(const float* __restrict__ nu, const float* __restrict__ th,
                 const float* __restrict__ dt, float* __restrict__ A) {
  int i = blockIdx.x * blockDim.x + threadIdx.x;
  if (i >= NA_) return;
  float d = dt[i / (RC_ * W_)];          // index (b*L + l), cached in scalar/L0
  float decay = __expf(-nu[i] * d);
  float s, c;
  __sincosf(th[i] * d, &s, &c);
  A[i]       = decay * c;                // Ar plane
  A[NA_ + i] = decay * s;                // Ai plane
}

// Shared serial-scan body: LDS holds Ar[L_*W_] then Ai[L_*W_] (planar,
// conflict-free: consecutive lanes hit consecutive words).
__device__ __forceinline__ void
scan_body(const float* __restrict__ As, int b, int rc,
          const float* __restrict__ ur, const float* __restrict__ ui,
          float* __restrict__ out) {
  const int base0 = b * L_ * RC_ + rc;
  const int hw0 = blockIdx.y * HW_CHUNK_;
  for (int hw = hw0 + (int)threadIdx.x; hw < hw0 + HW_CHUNK_; hw += 256) {
    const int w = hw % W_;
    size_t idx = (size_t)base0 * HW_ + (size_t)hw;
    float hr = 0.0f, hi = 0.0f;
#pragma unroll
    for (int l = 0; l < L_; ++l) {
      const float ar = As[l * W_ + w];
      const float ai = As[LW_ + l * W_ + w];
      const float vr = __builtin_nontemporal_load(&ur[idx]);
      const float vi = __builtin_nontemporal_load(&ui[idx]);
      const float nr = fmaf(ar, hr, fmaf(-ai, hi, vr));
      const float ni = fmaf(ar, hi, fmaf( ai, hr, vi));
      hr = nr; hi = ni;
      __builtin_nontemporal_store(hr, &out[idx]);
      __builtin_nontemporal_store(hi, &out[NTOT_ + idx]);
      idx += LSTRIDE_;
    }
  }
}

// ---------------------------------------------------------------------------
// Phase 2 (TDM path): one block per (b, rc) x H-chunk. Wave 0 issues two
// tensor_load_to_lds DMAs (Ar tile, Ai tile): 2D tile, 24 rows x 49 fp32,
// row stride RC_*W_ elements -> packed contiguously into LDS. Then all 8
// waves stream the recurrence with NT loads/stores (16x LDS reuse of A).
// ---------------------------------------------------------------------------
__global__ __launch_bounds__(256) void
scan_kernel_tdm(const float* __restrict__ Aws,
                const float* __restrict__ ur, const float* __restrict__ ui,
                float* __restrict__ out) {
  __shared__ float As[2 * LW_];          // 9408 bytes
  const int brc = blockIdx.x;            // b*RC + rc
  const int b = brc / RC_, rc = brc % RC_;

#if HAVE_TDM
  if (threadIdx.x < 32) {                // one TDM issue per block (wave 0)
    const uint32_t lds_base = (uint32_t)(uintptr_t)(&As[0]); // low 32b = LDS addr
    const uint64_t g_r = (uint64_t)(uintptr_t)Aws +
                         4ull * (uint64_t)W_ * (uint64_t)(b * L_ * RC_ + rc);
    const uint64_t g_i = g_r + 4ull * (uint64_t)NA_;

    i32x8 g1;
    g1[0] = (int)(2u << 16);   // workgroup_mask=0 (not in cluster), data_size=4B
    g1[1] = (int)((uint32_t)W_ << 16);   // tensor_dim0 = 49
    g1[2] = (int)((uint32_t)L_ << 16);   // tensor_dim1 = 24
    g1[3] = (int)((uint32_t)W_ << 16);   // tile_dim0   = 49
    g1[4] = L_;                          // tile_dim1 = 24, tile_dim2 = 0
    g1[5] = RC_ * W_;                    // tensor_dim0_stride = 12544 elements
    g1[6] = 0;
    g1[7] = 0;
    i32x4 g2 = (i32x4)0, g3 = (i32x4)0;  // unused dims (tile_dim3/4 = 0)
    i32x8 g4 = (i32x8)0;                 // trailing group (6-arg form), unused

    u32x4 g0r;
    g0r[0] = 1u;                               // count=1, is_restore=0
    g0r[1] = lds_base;                         // LDS dest: Ar at offset 0
    g0r[2] = (uint32_t)g_r;                    // global_addr[31:0]
    g0r[3] = (uint32_t)((g_r >> 32) & 0x1FFFFFFu) | (2u << 30); // addr[56:32], type=2
    __builtin_amdgcn_tensor_load_to_lds(g0r, g1, g2, g3, g4, 0);

    u32x4 g0i;
    g0i[0] = 1u;
    g0i[1] = lds_base + 4u * LW_;              // LDS dest: Ai plane
    g0i[2] = (uint32_t)g_i;
    g0i[3] = (uint32_t)((g_i >> 32) & 0x1FFFFFFu) | (2u << 30);
    __builtin_amdgcn_tensor_load_to_lds(g0i, g1, g2, g3, g4, 0);

    __builtin_amdgcn_s_wait_tensorcnt(0);      // TENSORcnt == 0: both tiles in LDS
  }
#else
  for (int e = threadIdx.x; e < LW_; e += 256) {
    const int src = (b * L_ + (e / W_)) * RC_ * W_ + rc * W_ + (e % W_);
    As[e]       = Aws[src];
    As[LW_ + e] = Aws[NA_ + src];
  }
#endif
  __syncthreads();
  scan_body(As, b, rc, ur, ui, out);
}

// ---------------------------------------------------------------------------
// Fallback (workspace too small): compute A directly into LDS per block.
// ---------------------------------------------------------------------------
__global__ __launch_bounds__(256) void
scan_kernel_fused(const float* __restrict__ nu, const float* __restrict__ th,
                  const float* __restrict__ dtp,
                  const float* __restrict__ ur, const float* __restrict__ ui,
                  float* __restrict__ out) {
  __shared__ float As[2 * LW_];
  const int brc = blockIdx.x;
  const int b = brc / RC_, rc = brc % RC_;
  for (int e = threadIdx.x; e < LW_; e += 256) {
    const int l = e / W_, w = e % W_;
    const int i = (b * L_ + l) * RC_ * W_ + rc * W_ + w;
    const float d = dtp[b * L_ + l];
    const float decay = __expf(-nu[i] * d);
    float s, c;
    __sincosf(th[i] * d, &s, &c);
    As[e]       = decay * c;
    As[LW_ + e] = decay * s;
  }
  __syncthreads();
  scan_body(As, b, rc, ur, ui, out);
}

extern "C" void kernel_launch(void* const* d_in, const int* in_sizes, int n_in,
                              void* d_out, int out_size, void* d_ws, size_t ws_size,
                              hipStream_t stream) {
  const float* nu = (const float*)d_in[0];   // (B,L,R,C,Wf)
  const float* th = (const float*)d_in[1];   // (B,L,R,C,Wf)
  const float* dt = (const float*)d_in[2];   // (B,L)
  const float* ur = (const float*)d_in[3];   // (B,L,R,C,H,Wf)
  const float* ui = (const float*)d_in[4];   // (B,L,R,C,H,Wf)
  float* out = (float*)d_out;                // (2,B,L,R,C,H,Wf) fp32
  float* Aws = (float*)d_ws;                 // planar Ar[NA_], Ai[NA_]

  const size_t ws_needed = (size_t)2 * NA_ * sizeof(float); // ~4.8 MB
  dim3 grid(B_ * RC_, HW_ / HW_CHUNK_);      // 512 x 3 blocks, 8 waves each

  if (ws_size >= ws_needed) {
    compute_A_kernel<<<(NA_ + 255) / 256, 256, 0, stream>>>(nu, th, dt, Aws);
    scan_kernel_tdm<<<grid, 256, 0, stream>>>(Aws, ur, ui, out);
  } else {
    scan_kernel_fused<<<grid, 256, 0, stream>>>(nu, th, dt, ur, ui, out);
  }
}